// BiLSTM_CRF_25400436588898
// MI455X (gfx1250) — compile-verified
//
#include <hip/hip_runtime.h>
#include <hip/hip_bf16.h>
#include <math.h>

// Problem constants (match reference)
#define TT 512
#define BB 128
#define HH 128      // per-direction hidden
#define EE 128      // embedding dim
#define KK 17       // CRF states

typedef float v2f __attribute__((ext_vector_type(2)));
typedef float v8f __attribute__((ext_vector_type(8)));

#define XS_STRIDE 132           // 132 % 64 banks = 4 -> conflict-free column reads
#define PK_LSTM   (8 * 32 * 4 * 64)   // per-matrix packed fragment count = 65536
#define PK_EMIT   (2 * 64 * 64)       // 8192

// -------------------------------------------------------------------------
// Weight packing: rearrange W[n][k] into per-wave WMMA B-fragment order
//   dst[((jt*32 + kt)*4 + g)*64 + lane*2 + c] = W[n*128 + k]
//   n = g*128 + jt*16 + (lane&15),  k = kt*4 + ((lane>>4)<<1) + c
// so the hot loop fetches each fragment as one contiguous 8B/lane access.
// -------------------------------------------------------------------------
__global__ __launch_bounds__(256)
void pack_lstm_kernel(const float* __restrict__ w, float* __restrict__ dst)
{
    int e = blockIdx.x * 256 + threadIdx.x;          // 0 .. 65535
    int c  = e & 1;
    int l  = (e >> 1) & 31;
    int g  = (e >> 6) & 3;
    int kt = (e >> 8) & 31;
    int jt = (e >> 13) & 7;
    int n  = g * HH + jt * 16 + (l & 15);
    int k  = kt * 4 + ((l >> 4) << 1) + c;
    dst[e] = w[n * 128 + k];
}

__global__ __launch_bounds__(256)
void pack_emit_kernel(const float* __restrict__ w_emit, float* __restrict__ dst)
{
    int e = blockIdx.x * 256 + threadIdx.x;          // 0 .. 8191
    int c  = e & 1;
    int l  = (e >> 1) & 31;
    int kt = (e >> 6) & 63;
    int nt = (e >> 12) & 1;
    int n  = nt * 16 + (l & 15);
    int k  = kt * 4 + ((l >> 4) << 1) + c;
    dst[e] = (n < KK) ? w_emit[n * (2 * HH) + k] : 0.0f;
}

// -------------------------------------------------------------------------
// Fused embedding-gather + input-projection + recurrent BiLSTM.
// grid = (8 batch-tiles, 2 directions), block = 256 (8 waves).
// Wave jt owns the four gate tiles (i,f,g,o) of one 16x16 (batch,hidden)
// block: cell state c lives in a v8f register (D-matrix layout) all sequence.
// w_ih fragments live in 256 VGPRs (loaded once); w_hh fragments live in
// 256KB of LDS (loaded once); the 512-step loop touches only LDS+registers.
// -------------------------------------------------------------------------
__global__ __launch_bounds__(256)
void bilstm_kernel(const int* __restrict__ ids,
                   const float* __restrict__ embed,
                   const float* __restrict__ pk_ih_f, const float* __restrict__ pk_hh_f,
                   const float* __restrict__ b_ih_f,  const float* __restrict__ b_hh_f,
                   const float* __restrict__ pk_ih_b, const float* __restrict__ pk_hh_b,
                   const float* __restrict__ b_ih_b,  const float* __restrict__ b_hh_b,
                   float* __restrict__ hs_f, float* __restrict__ hs_b)
{
    extern __shared__ float smem[];
    float* xs  = smem;                       // [16][XS_STRIDE] x_t tile
    float* hsh = smem + 16 * XS_STRIDE;      // [16][XS_STRIDE] h_{t-1} tile
    float* wl  = smem + 32 * XS_STRIDE;      // [PK_LSTM] packed w_hh fragments

    const int tid  = threadIdx.x;
    const int lane = tid & 31;
    const int wave = tid >> 5;
    const int mb   = blockIdx.x;             // batch tile 0..7
    const int dir  = blockIdx.y;             // 0 fwd, 1 bwd

    const float* pk_ih = dir ? pk_ih_b : pk_ih_f;
    const float* pk_hh = dir ? pk_hh_b : pk_hh_f;
    const float* bih   = dir ? b_ih_b  : b_ih_f;
    const float* bhh   = dir ? b_hh_b  : b_hh_f;
    float*       hout  = dir ? hs_b    : hs_f;

    const int jt    = wave;                  // hidden column tile 0..7
    const int rowa  = lane & 15;             // A-frag row / D col
    const int koff  = (lane >> 4) * 2;       // K sub-offset per half-wave
    const int hirow = (lane >> 4) << 3;      // +8 D-row offset in hi half

    // ---- one-time setup --------------------------------------------------
    // w_hh fragments -> LDS (whole 256KB, cooperative float4 copy)
    for (int i = tid * 4; i < PK_LSTM; i += 256 * 4)
        *(float4*)&wl[i] = *(const float4*)&pk_hh[i];

    // w_ih fragments -> registers (this wave's 32KB slice)
    v2f wr[32][4];
    {
        const float* mypk = pk_ih + (jt * 32 * 4) * 64 + lane * 2;
#pragma unroll
        for (int kt = 0; kt < 32; ++kt)
#pragma unroll
            for (int g = 0; g < 4; ++g)
                wr[kt][g] = *(const v2f*)(mypk + (kt * 4 + g) * 64);
    }

    // zero h state
    for (int i = tid; i < 16 * XS_STRIDE; i += 256) hsh[i] = 0.0f;

    v8f cst = {0.f, 0.f, 0.f, 0.f, 0.f, 0.f, 0.f, 0.f};
    const v8f zero8 = {0.f, 0.f, 0.f, 0.f, 0.f, 0.f, 0.f, 0.f};

    float bias[4];
#pragma unroll
    for (int g = 0; g < 4; ++g) {
        int n = g * HH + jt * 16 + rowa;
        bias[g] = bih[n] + bhh[n];
    }

    const float* mywl = wl + (jt * 32 * 4) * 64 + lane * 2;  // this wave's hh slice

    __syncthreads();

    // ---- recurrence ------------------------------------------------------
    for (int step = 0; step < TT; ++step) {
        const int t = dir ? (TT - 1 - step) : step;

        // gather x_t = embed[ids[b,t]] into LDS (16 rows x 128, coalesced)
        {
            const int r   = tid >> 4;
            const int seg = tid & 15;
            const int gid = ids[(mb * 16 + r) * TT + t];
            const float* src = embed + (size_t)gid * EE + seg * 8;
            *(float4*)&xs[r * XS_STRIDE + seg * 8]     = *(const float4*)(src);
            *(float4*)&xs[r * XS_STRIDE + seg * 8 + 4] = *(const float4*)(src + 4);
        }
        __syncthreads();

        v8f acc[4] = {zero8, zero8, zero8, zero8};

        // x half: A from xs LDS, B from registers
#pragma unroll
        for (int kt = 0; kt < 32; ++kt) {
            v2f a = *(const v2f*)&xs[rowa * XS_STRIDE + kt * 4 + koff];
#pragma unroll
            for (int g = 0; g < 4; ++g)
                acc[g] = __builtin_amdgcn_wmma_f32_16x16x4_f32(
                    false, a, false, wr[kt][g], (short)0, acc[g], false, false);
        }
        // h half: A from hsh LDS, B from wl LDS (conflict-free 8B/lane)
#pragma unroll
        for (int kt = 0; kt < 32; ++kt) {
            v2f a = *(const v2f*)&hsh[rowa * XS_STRIDE + kt * 4 + koff];
#pragma unroll
            for (int g = 0; g < 4; ++g) {
                v2f b = *(const v2f*)(mywl + (kt * 4 + g) * 64);
                acc[g] = __builtin_amdgcn_wmma_f32_16x16x4_f32(
                    false, a, false, b, (short)0, acc[g], false, false);
            }
        }
        __syncthreads();  // all LDS reads done before h overwrite

        // elementwise LSTM cell; write h to LDS and to global hs[t][b][j]
        const int jcol = jt * 16 + rowa;
#pragma unroll
        for (int v = 0; v < 8; ++v) {
            const int lrow = v + hirow;
            float iv = acc[0][v] + bias[0];
            float fv = acc[1][v] + bias[1];
            float gv = acc[2][v] + bias[2];
            float ov = acc[3][v] + bias[3];
            float si = 1.0f / (1.0f + __expf(-iv));
            float sf = 1.0f / (1.0f + __expf(-fv));
            float so = 1.0f / (1.0f + __expf(-ov));
            float c  = sf * cst[v] + si * tanhf(gv);
            cst[v] = c;
            float h = so * tanhf(c);
            hsh[lrow * XS_STRIDE + jcol] = h;
            hout[((size_t)t * BB + (mb * 16 + lrow)) * HH + jcol] = h;
        }
        __syncthreads();
    }
}

// -------------------------------------------------------------------------
// Emissions: [T*B, 256] @ w_emit^T(256,17) + b_emit, K padded to 32.
// Packed w_emit fragments staged once in LDS; the 16x256 A-slab is prefetched
// into a 64-entry v2f register array so all global loads pipeline.
// -------------------------------------------------------------------------
__global__ __launch_bounds__(256)
void emit_kernel(const float* __restrict__ hs_f, const float* __restrict__ hs_b,
                 const float* __restrict__ pk_em, const float* __restrict__ b_emit,
                 float* __restrict__ em)
{
    __shared__ float wlE[PK_EMIT];   // 32KB
    const int tid  = threadIdx.x;
    const int lane = tid & 31;
    const int wave = tid >> 5;

    for (int i = tid * 4; i < PK_EMIT; i += 256 * 4)
        *(float4*)&wlE[i] = *(const float4*)&pk_em[i];
    __syncthreads();

    const int mt   = blockIdx.x * 8 + wave;      // 16-row tile index
    const size_t rbase = (size_t)mt * 16;
    const int rowa  = lane & 15;
    const int koff  = (lane >> 4) * 2;
    const int hirow = (lane >> 4) << 3;

    // prefetch A fragments (pipelined global loads, no per-wmma waits)
    v2f af[64];
    {
        const float* pf = hs_f + (rbase + rowa) * HH + koff;
        const float* pb = hs_b + (rbase + rowa) * HH + koff;
#pragma unroll
        for (int kt = 0; kt < 32; ++kt) af[kt]      = *(const v2f*)(pf + kt * 4);
#pragma unroll
        for (int kt = 0; kt < 32; ++kt) af[kt + 32] = *(const v2f*)(pb + kt * 4);
    }

    const v8f zero8 = {0.f, 0.f, 0.f, 0.f, 0.f, 0.f, 0.f, 0.f};
    v8f acc0 = zero8, acc1 = zero8;
    const float* myw = wlE + lane * 2;
#pragma unroll
    for (int kt = 0; kt < 64; ++kt) {
        v2f b0 = *(const v2f*)(myw + kt * 64);
        v2f b1 = *(const v2f*)(myw + (64 + kt) * 64);
        acc0 = __builtin_amdgcn_wmma_f32_16x16x4_f32(false, af[kt], false, b0,
                                                     (short)0, acc0, false, false);
        acc1 = __builtin_amdgcn_wmma_f32_16x16x4_f32(false, af[kt], false, b1,
                                                     (short)0, acc1, false, false);
    }

    const int col = lane & 15;
    const float be0  = b_emit[col];
    const float be16 = b_emit[16];
#pragma unroll
    for (int v = 0; v < 8; ++v) {
        const size_t r = rbase + v + hirow;
        em[r * KK + col] = acc0[v] + be0;
        if (col == 0) em[r * KK + 16] = acc1[v] + be16;
    }
}

// -------------------------------------------------------------------------
// CRF NLL: one wave per batch element; alpha in lanes 0..16, cross-lane
// logsumexp via shuffles; gold-path score on the side; mean via atomics.
// -------------------------------------------------------------------------
__global__ __launch_bounds__(32)
void crf_kernel(const float* __restrict__ em, const int* __restrict__ tags,
                const unsigned char* __restrict__ mask,
                const float* __restrict__ start_tr, const float* __restrict__ end_tr,
                const float* __restrict__ trans, float* __restrict__ out)
{
    const int b    = blockIdx.x;
    const int lane = threadIdx.x;
    __shared__ float trS[KK * KK];
    __shared__ float stS[KK], etS[KK];
    for (int i = lane; i < KK * KK; i += 32) trS[i] = trans[i];
    if (lane < KK) { stS[lane] = start_tr[lane]; etS[lane] = end_tr[lane]; }
    __syncthreads();

    const float NEG = -1e30f;
    const int  j    = lane;
    const bool actv = (j < KK);

    float alpha = actv ? (stS[j] + em[(size_t)b * KK + j]) : NEG;
    int   tg    = tags[b * TT];
    float score = stS[tg] + em[(size_t)b * KK + tg];
    int   prev  = tg;

    for (int t = 1; t < TT; ++t) {
        const size_t ebase = ((size_t)t * BB + b) * KK;
        const bool m = (mask[b * TT + t] != 0);

        const int tgt = tags[b * TT + t];
        if (m) { score += trS[prev * KK + tgt] + em[ebase + tgt]; prev = tgt; }

        float mx = NEG;
#pragma unroll
        for (int i = 0; i < KK; ++i) {
            float ai = __shfl(alpha, i, 32);
            float v  = ai + (actv ? trS[i * KK + j] : 0.0f);
            mx = fmaxf(mx, v);
        }
        float s = 0.0f;
#pragma unroll
        for (int i = 0; i < KK; ++i) {
            float ai = __shfl(alpha, i, 32);
            float v  = ai + (actv ? trS[i * KK + j] : 0.0f);
            s += __expf(v - mx);
        }
        float nxt = mx + __logf(s) + (actv ? em[ebase + j] : 0.0f);
        if (actv && m) alpha = nxt;
    }
    score += etS[prev];

    float val = actv ? (alpha + etS[j]) : NEG;
    float mx = val;
    for (int off = 16; off > 0; off >>= 1) mx = fmaxf(mx, __shfl_xor(mx, off, 32));
    float s = __expf(val - mx);
    for (int off = 16; off > 0; off >>= 1) s += __shfl_xor(s, off, 32);
    float logZ = mx + __logf(s);

    if (lane == 0) atomicAdd(out, -(score - logZ) * (1.0f / (float)BB));
}

__global__ void zero_out_kernel(float* o)
{
    if (threadIdx.x == 0 && blockIdx.x == 0) o[0] = 0.0f;
}

extern "C" void kernel_launch(void* const* d_in, const int* in_sizes, int n_in,
                              void* d_out, int out_size, void* d_ws, size_t ws_size,
                              hipStream_t stream)
{
    (void)in_sizes; (void)n_in; (void)out_size; (void)ws_size;

    const int*   ids     = (const int*)d_in[0];
    const int*   tags    = (const int*)d_in[1];
    const unsigned char* mask = (const unsigned char*)d_in[2];
    const float* embed   = (const float*)d_in[3];
    const float* w_ih_f  = (const float*)d_in[4];
    const float* w_hh_f  = (const float*)d_in[5];
    const float* b_ih_f  = (const float*)d_in[6];
    const float* b_hh_f  = (const float*)d_in[7];
    const float* w_ih_b  = (const float*)d_in[8];
    const float* w_hh_b  = (const float*)d_in[9];
    const float* b_ih_b  = (const float*)d_in[10];
    const float* b_hh_b  = (const float*)d_in[11];
    const float* w_emit  = (const float*)d_in[12];
    const float* b_emit  = (const float*)d_in[13];
    const float* st      = (const float*)d_in[14];
    const float* et      = (const float*)d_in[15];
    const float* trans   = (const float*)d_in[16];
    float*       out     = (float*)d_out;

    // workspace: hs_f | hs_b | emissions | 4x packed LSTM weights | packed emit
    float* ws      = (float*)d_ws;
    float* hs_f    = ws;
    float* hs_b    = hs_f + (size_t)TT * BB * HH;
    float* em      = hs_b + (size_t)TT * BB * HH;
    float* pk_ih_f = em + (size_t)TT * BB * KK;
    float* pk_hh_f = pk_ih_f + PK_LSTM;
    float* pk_ih_b = pk_hh_f + PK_LSTM;
    float* pk_hh_b = pk_ih_b + PK_LSTM;
    float* pk_em   = pk_hh_b + PK_LSTM;

    zero_out_kernel<<<1, 32, 0, stream>>>(out);

    pack_lstm_kernel<<<PK_LSTM / 256, 256, 0, stream>>>(w_ih_f, pk_ih_f);
    pack_lstm_kernel<<<PK_LSTM / 256, 256, 0, stream>>>(w_hh_f, pk_hh_f);
    pack_lstm_kernel<<<PK_LSTM / 256, 256, 0, stream>>>(w_ih_b, pk_ih_b);
    pack_lstm_kernel<<<PK_LSTM / 256, 256, 0, stream>>>(w_hh_b, pk_hh_b);
    pack_emit_kernel<<<PK_EMIT / 256, 256, 0, stream>>>(w_emit, pk_em);

    dim3 g(8, 2);  // 8 batch tiles x 2 directions, persistent over T
    const size_t lds_bytes = (size_t)(32 * XS_STRIDE + PK_LSTM) * sizeof(float); // ~273KB
    bilstm_kernel<<<g, 256, lds_bytes, stream>>>(ids, embed,
                                                 pk_ih_f, pk_hh_f, b_ih_f, b_hh_f,
                                                 pk_ih_b, pk_hh_b, b_ih_b, b_hh_b,
                                                 hs_f, hs_b);

    emit_kernel<<<512, 256, 0, stream>>>(hs_f, hs_b, pk_em, b_emit, em);

    crf_kernel<<<BB, 32, 0, stream>>>(em, tags, mask, st, et, trans, out);
}